// PartAttention_58755152609478
// MI455X (gfx1250) — compile-verified
//
#include <hip/hip_runtime.h>
#include <hip/hip_bf16.h>
#include <math.h>

typedef __attribute__((ext_vector_type(2))) float v2f;
typedef __attribute__((ext_vector_type(8))) float v8f;
typedef int v4i_ __attribute__((vector_size(16)));   // 16B unit for async DMA

#define B_   32
#define C_   256
#define HW_  4096
#define P_   8
#define NSPLIT 8          // hw partitions for GEMM2
#define KC   16           // hw chunk staged in LDS per iteration
#define XSTRIDE (KC + 4)  // 20: padded LDS row stride (dwords) -> conflict-free

// workspace layout (float offsets)
#define SCORES_OFF 0                        // [B*P*HW]  = 1,048,576 f
#define MX_OFF     (B_ * P_ * HW_)          // [256]
#define SCALE_OFF  (MX_OFF + 256)           // [256]
#define PART_OFF   (SCALE_OFF + 256)        // [NSPLIT*B*P*C] = 524,288 f

#ifndef __has_builtin
#define __has_builtin(x) 0
#endif
#if __has_builtin(__builtin_amdgcn_global_load_async_to_lds_b128)
#define ASYNC_BUILTIN 1
#else
#define ASYNC_BUILTIN 0
#endif

// 16B global -> LDS async copy (ASYNCcnt-tracked DMA, no VGPR round trip)
__device__ __forceinline__ void async_copy16(const float* gp, float* lp)
{
#if ASYNC_BUILTIN
    __builtin_amdgcn_global_load_async_to_lds_b128(
        (__attribute__((address_space(1))) v4i_*)gp,
        (__attribute__((address_space(3))) v4i_*)lp, 0, 0);
#else
    const unsigned lds_off =
        (unsigned)(unsigned long long)(__attribute__((address_space(3))) const void*)lp;
    const unsigned long long ga = (unsigned long long)gp;
    asm volatile("global_load_async_to_lds_b128 %0, %1, off"
                 :: "v"(lds_off), "v"(ga) : "memory");
#endif
}

__device__ __forceinline__ void wait_asynccnt0()
{
    asm volatile("s_wait_asynccnt 0x0" ::: "memory");
}

// ---------------------------------------------------------------------------
// Kernel 1: scores[b,p,hw] = sum_c W[p,c]*x[b,c,hw] + bias[p]
// grid (HW/128, B), block 256 (8 waves); wave w owns a 16-wide hw tile.
// x is streamed straight into WMMA B operands (each element used once).
// ---------------------------------------------------------------------------
__global__ __launch_bounds__(256) void k1_scores(const float* __restrict__ x,
                                                 const float* __restrict__ Wm,
                                                 const float* __restrict__ bias,
                                                 float* __restrict__ scores)
{
    __shared__ float Wt[16 * 260];   // W padded to 16 rows, stride 260 (16.6 KB)
    const int b   = blockIdx.y;
    const int tid = threadIdx.x;

    for (int i = tid; i < 16 * C_; i += 256) {
        const int p = i >> 8;        // 0..15
        const int c = i & 255;       // coalesced over c
        Wt[p * 260 + c] = (p < P_) ? Wm[p * C_ + c] : 0.0f;
    }
    __syncthreads();

    const int wave  = tid >> 5;
    const int lane  = tid & 31;
    const int m     = lane & 15;           // M (A) / N (B) index within tile
    const int khalf = (lane >> 4) << 1;    // 0 for lanes 0-15, 2 for 16-31
    const int n0    = blockIdx.x * 128 + wave * 16;
    const float* xb = x + (size_t)b * C_ * HW_;

    v8f acc = {};
    #pragma unroll 8
    for (int k0 = 0; k0 < C_; k0 += 4) {
        // A = W-tile (16x4) from LDS: contiguous b64, banks 4m+k -> conflict-free
        const v2f a = *(const v2f*)&Wt[m * 260 + k0 + khalf];
        // B = x rows k0+khalf, k0+khalf+1 at hw = n0+m: two coalesced 64B segments
        v2f bv;
        bv.x = xb[(size_t)(k0 + khalf)     * HW_ + n0 + m];
        bv.y = xb[(size_t)(k0 + khalf + 1) * HW_ + n0 + m];
        acc = __builtin_amdgcn_wmma_f32_16x16x4_f32(false, a, false, bv,
                                                    (short)0, acc, false, false);
    }

    // D: vgpr v -> M=v (lanes 0-15). Rows 8..15 are zero-padding -> drop.
    if (lane < 16) {
        #pragma unroll
        for (int v = 0; v < P_; ++v)
            scores[((size_t)b * P_ + v) * HW_ + n0 + lane] = acc[v] + bias[v];
    }
}

// ---------------------------------------------------------------------------
// Kernel 2: per-(b,p) softmax stats: max and 1/(sum(exp)*HW)
// grid (B*P), block 256
// ---------------------------------------------------------------------------
__global__ __launch_bounds__(256) void k2_stats(const float* __restrict__ scores,
                                                float* __restrict__ mx,
                                                float* __restrict__ scale)
{
    __shared__ float red[256];
    const int bp  = blockIdx.x;
    const int tid = threadIdx.x;
    const float* row = scores + (size_t)bp * HW_;

    float lm = -3.402823466e38f;
    for (int i = tid; i < HW_; i += 256) lm = fmaxf(lm, row[i]);
    red[tid] = lm;
    __syncthreads();
    for (int s = 128; s > 0; s >>= 1) {
        if (tid < s) red[tid] = fmaxf(red[tid], red[tid + s]);
        __syncthreads();
    }
    const float mv = red[0];
    __syncthreads();

    float ls = 0.0f;
    for (int i = tid; i < HW_; i += 256) ls += __expf(row[i] - mv);
    red[tid] = ls;
    __syncthreads();
    for (int s = 128; s > 0; s >>= 1) {
        if (tid < s) red[tid] += red[tid + s];
        __syncthreads();
    }
    if (tid == 0) { mx[bp] = mv; scale[bp] = 1.0f / (red[0] * (float)HW_); }
}

// ---------------------------------------------------------------------------
// Kernel 3: partial[part,b,p,c] = sum_{hw in part} x[b,c,hw]*softmax_map[b,p,hw]
// grid (NSPLIT, B), block 256 (8 waves). Wave w owns c-tiles [32w,32w+32).
// X tiles DMA'd global->LDS asynchronously, double-buffered so HBM/L2 traffic
// overlaps the WMMA work on the other buffer.
// ---------------------------------------------------------------------------
__global__ __launch_bounds__(256) void k3_pool(const float* __restrict__ x,
                                               const float* __restrict__ scores,
                                               const float* __restrict__ mx,
                                               const float* __restrict__ scale,
                                               float* __restrict__ partial)
{
    __shared__ float Xs[2][C_ * XSTRIDE];   // 2 x 20,480 B
    __shared__ float Ms[16 * XSTRIDE];      // 1,280 B (single buffer: written and
                                            // read strictly between the 2 barriers)

    const int b    = blockIdx.y;
    const int part = blockIdx.x;
    const int tid  = threadIdx.x;
    const int wave = tid >> 5;
    const int lane = tid & 31;
    const int m     = lane & 15;
    const int khalf = (lane >> 4) << 1;
    const int c0    = wave * 32;

    const float* xb = x + (size_t)b * C_ * HW_;

    // cooperative X loader role: 4 lanes x 16B per 16-float row, 64 rows/pass
    const int cl  = tid >> 2;        // 0..63
    const int off = (tid & 3) * 4;   // 0,4,8,12

    // softmax-map producer role: thread tid owns map element (p = tid>>4, k = tid&15)
    const int pM = tid >> 4;         // 0..15 (rows 8..15 are zero padding)
    const int kM = tid & 15;
    float rowmax = 0.0f, rowscale = 0.0f;
    const float* srow = scores + ((size_t)b * P_ + (pM & 7)) * HW_;
    if (pM < P_) {
        rowmax   = mx[b * P_ + pM];
        rowscale = scale[b * P_ + pM];
    }

    auto issue_chunk = [&](int buf, int hw0) {
        #pragma unroll
        for (int r = 0; r < 4; ++r) {
            const int c = cl + r * 64;
            async_copy16(&xb[(size_t)c * HW_ + hw0 + off],
                         &Xs[buf][c * XSTRIDE + off]);
        }
    };

    v8f acc0 = {}, acc1 = {};

    const int HWPART = HW_ / NSPLIT;     // 512
    const int NCH    = HWPART / KC;      // 32
    const int hwbase = part * HWPART;

    issue_chunk(0, hwbase);              // prologue prefetch

    for (int chunk = 0; chunk < NCH; ++chunk) {
        const int cur = chunk & 1;
        const int hw0 = hwbase + chunk * KC;

        wait_asynccnt0();                // my DMA into Xs[cur] landed
        // softmax map tile for this chunk
        float mval = 0.0f;
        if (pM < P_) mval = __expf(srow[hw0 + kM] - rowmax) * rowscale;
        Ms[pM * XSTRIDE + kM] = mval;
        __syncthreads();                 // everyone's DMA + map stores visible

        if (chunk + 1 < NCH)             // overlap next chunk's DMA with WMMAs
            issue_chunk(cur ^ 1, hw0 + KC);

        #pragma unroll
        for (int k0 = 0; k0 < KC; k0 += 4) {
            const v2f bv = *(const v2f*)&Ms[m * XSTRIDE + k0 + khalf];
            const v2f a0 = *(const v2f*)&Xs[cur][(c0 + m)      * XSTRIDE + k0 + khalf];
            const v2f a1 = *(const v2f*)&Xs[cur][(c0 + 16 + m) * XSTRIDE + k0 + khalf];
            acc0 = __builtin_amdgcn_wmma_f32_16x16x4_f32(false, a0, false, bv,
                                                         (short)0, acc0, false, false);
            acc1 = __builtin_amdgcn_wmma_f32_16x16x4_f32(false, a1, false, bv,
                                                         (short)0, acc1, false, false);
        }
        __syncthreads();                 // Xs[cur] reads done before refill at chunk+2
    }

    // D layout: vgpr v -> (c = base + [hi?8:0] + v, p = lane&15); p>=8 is padding
    float* pp = partial + ((size_t)part * B_ + b) * (P_ * C_);
    const int p = lane & 15;
    if (p < P_) {
        const int chi = (lane >= 16) ? 8 : 0;
        #pragma unroll
        for (int v = 0; v < 8; ++v) {
            pp[p * C_ + c0 +      chi + v] = acc0[v];
            pp[p * C_ + c0 + 16 + chi + v] = acc1[v];
        }
    }
}

// ---------------------------------------------------------------------------
// Kernel 4: out[i] = sum_part partial[part][i]   (deterministic reduction)
// ---------------------------------------------------------------------------
__global__ __launch_bounds__(256) void k4_reduce(const float* __restrict__ partial,
                                                 float* __restrict__ out)
{
    const int i = blockIdx.x * 256 + threadIdx.x;
    float s = 0.0f;
    #pragma unroll
    for (int r = 0; r < NSPLIT; ++r)
        s += partial[(size_t)r * (B_ * P_ * C_) + i];
    out[i] = s;
}

// ---------------------------------------------------------------------------
extern "C" void kernel_launch(void* const* d_in, const int* in_sizes, int n_in,
                              void* d_out, int out_size, void* d_ws, size_t ws_size,
                              hipStream_t stream)
{
    const float* x    = (const float*)d_in[0];   // [32,256,64,64]
    const float* Wm   = (const float*)d_in[1];   // [8,256]
    const float* bias = (const float*)d_in[2];   // [8]
    float* out = (float*)d_out;                  // [32, 2048]

    float* ws      = (float*)d_ws;
    float* scores  = ws + SCORES_OFF;
    float* mx      = ws + MX_OFF;
    float* scale   = ws + SCALE_OFF;
    float* partial = ws + PART_OFF;

    k1_scores<<<dim3(HW_ / 128, B_), 256, 0, stream>>>(x, Wm, bias, scores);
    k2_stats <<<dim3(B_ * P_),       256, 0, stream>>>(scores, mx, scale);
    k3_pool  <<<dim3(NSPLIT, B_),    256, 0, stream>>>(x, scores, mx, scale, partial);
    k4_reduce<<<dim3((B_ * P_ * C_) / 256), 256, 0, stream>>>(partial, out);
}